// DecoderRNN_88923002896563
// MI455X (gfx1250) — compile-verified
//
#include <hip/hip_runtime.h>
#include <hip/hip_bf16.h>
#include <math.h>

// ---------------- problem constants (match reference) ----------------
#define BATCH   8
#define BEAM    4
#define BB      32          // BATCH*BEAM
#define TSTEPS  40
#define VOCAB   32000
#define EXT     32100
#define HDIM    512
#define ENC_H   512
#define D_E     256
#define EDIM    1024        // H + ENC_H
#define COMB    1024
#define SLEN    128
#define SOS     1
#define EOS     2
#define UNK     3
#define NEGV    (-1e30f)
#define EPSC    (1e-31f)

typedef __attribute__((ext_vector_type(2))) float v2f;
typedef __attribute__((ext_vector_type(8))) float v8f;

__device__ __forceinline__ float sigf(float x) { return 1.0f / (1.0f + expf(-x)); }

// ---------------- init: h, cov, score, x, closs ----------------
__global__ void k_init(float* h_cur, float* cov, int* x_tok, float* score,
                       float* closs, const float* __restrict__ init_h) {
    int i = blockIdx.x * blockDim.x + threadIdx.x;
    // h_cur[m][j] = init_h[0][m/4][j]   (repeat over beams)
    for (int idx = i; idx < BB * HDIM; idx += gridDim.x * blockDim.x) {
        int m = idx / HDIM, j = idx - m * HDIM;
        h_cur[idx] = init_h[(m >> 2) * HDIM + j];
    }
    for (int idx = i; idx < BB * SLEN; idx += gridDim.x * blockDim.x) cov[idx] = 0.0f;
    if (i < BB) {
        x_tok[i] = SOS;
        score[i] = ((i & 3) == 0) ? 0.0f : NEGV;
    }
    if (i == 0) *closs = 0.0f;
}

// ---------------- embedding gather with OOV filter ----------------
__global__ void k_embed(const int* __restrict__ x_tok,
                        const float* __restrict__ embedding,
                        float* __restrict__ emb) {
    int i = blockIdx.x * blockDim.x + threadIdx.x;      // BB*EDIM threads
    if (i >= BB * EDIM) return;
    int m = i / EDIM, e = i - m * EDIM;
    int tok = x_tok[m];
    if ((unsigned)tok >= (unsigned)VOCAB) tok = UNK;
    emb[i] = embedding[(size_t)tok * EDIM + e];
}

// ---------------- fp32 WMMA GEMM: D[M=32,N] = A[32,K] * op(B) + bias ----------------
// TRANSB=true : B stored [N,K] row-major (w_ih / w_hh / w_out), i.e. D = A * B^T
// TRANSB=false: B stored [K,N] row-major (enc_bl_w),            i.e. D = A * B
template <bool TRANSB>
__global__ void k_gemm_wmma(const float* __restrict__ A, const float* __restrict__ Bm,
                            const float* __restrict__ bias, float* __restrict__ D,
                            int N, int K, int lda, int ldb, int ldd) {
    const int wave = (blockIdx.x * blockDim.x + threadIdx.x) >> 5;   // 1 wave = 1 16x16 tile
    const int lane = threadIdx.x & 31;
    const int tileM = wave & 1;            // M=32 -> 2 M-tiles
    const int tileN = wave >> 1;
    if (tileN * 16 >= N) return;           // wave-uniform exit (EXEC all-ones inside)

    const int mrow  = tileM * 16 + (lane & 15);   // A row for this lane
    const int ncol  = tileN * 16 + (lane & 15);   // B/D column for this lane
    const int khalf = (lane >> 4) << 1;           // 0 for lanes 0-15, 2 for 16-31

    v8f acc = {};
    for (int k = 0; k < K; k += 4) {
        const int kk = k + khalf;
        // A 16x4 layout: VGPR0=K(kk), VGPR1=K(kk+1); contiguous -> one b64 load
        v2f a = *(const v2f*)(A + (size_t)mrow * lda + kk);
        v2f b;
        if constexpr (TRANSB) {
            b = *(const v2f*)(Bm + (size_t)ncol * ldb + kk);
            __builtin_prefetch(Bm + (size_t)ncol * ldb + kk + 64, 0, 1);
        } else {
            b.x = Bm[(size_t)kk * ldb + ncol];
            b.y = Bm[(size_t)(kk + 1) * ldb + ncol];
        }
        acc = __builtin_amdgcn_wmma_f32_16x16x4_f32(
            /*neg_a=*/false, a, /*neg_b=*/false, b,
            /*c_mod=*/(short)0, acc, /*reuse_a=*/false, /*reuse_b=*/false);
    }
    const float bv = bias ? bias[ncol] : 0.0f;
    const int rowbase = tileM * 16 + ((lane >> 4) << 3);   // rows v / v+8 split
#pragma unroll
    for (int v = 0; v < 8; ++v)
        D[(size_t)(rowbase + v) * ldd + ncol] = acc[v] + bv;
}

// ---------------- GRU elementwise combine (gate order r,z,n) ----------------
__global__ void k_gru_combine(const float* __restrict__ gi, const float* __restrict__ gh,
                              const float* __restrict__ h_cur,
                              float* __restrict__ h_new, float* __restrict__ combined) {
    int i = blockIdx.x * blockDim.x + threadIdx.x;    // BB*HDIM
    if (i >= BB * HDIM) return;
    int m = i / HDIM, j = i - m * HDIM;
    const float* gim = gi + (size_t)m * 3 * HDIM;
    const float* ghm = gh + (size_t)m * 3 * HDIM;
    float r = sigf(gim[j] + ghm[j]);
    float z = sigf(gim[HDIM + j] + ghm[HDIM + j]);
    float n = tanhf(gim[2 * HDIM + j] + r * ghm[2 * HDIM + j]);
    float hv = (1.0f - z) * n + z * h_cur[i];
    h_new[i] = hv;
    combined[(size_t)m * COMB + j] = hv;              // front half of [h_new, ctx]
}

// ---------------- attention + coverage + context (one block per row) ----------------
__global__ void k_attn(const float* __restrict__ hW,        // [BB,ENC_H] = h_new @ enc_bl_w
                       const float* __restrict__ enc,       // [B,S,ENC_H]
                       const float* __restrict__ enc_bl_b,
                       const float* __restrict__ cover_w,
                       float* __restrict__ cov, float* __restrict__ attn,
                       float* __restrict__ combined, float* __restrict__ closs, int t) {
    __shared__ float red[SLEN];
    __shared__ float sm[SLEN];
    const int m = blockIdx.x;
    const int s = threadIdx.x;                  // 128 threads
    const int b = m >> 2;
    const float* tm = hW + (size_t)m * ENC_H;
    const float* es = enc + ((size_t)b * SLEN + s) * ENC_H;

    float e = enc_bl_b[0];
    float dot = 0.0f;
    for (int f = 0; f < ENC_H; ++f) dot += tm[f] * es[f];
    e += dot;
    float cv = cov[m * SLEN + s];
    if (t > 0) e += cover_w[0] * logf(cv + EPSC);

    // softmax over S=128
    red[s] = e; __syncthreads();
    for (int off = 64; off > 0; off >>= 1) {
        if (s < off) red[s] = fmaxf(red[s], red[s + off]);
        __syncthreads();
    }
    float mx = red[0]; __syncthreads();
    float ex = expf(e - mx);
    red[s] = ex; __syncthreads();
    for (int off = 64; off > 0; off >>= 1) {
        if (s < off) red[s] += red[s + off];
        __syncthreads();
    }
    float a = ex / red[0]; __syncthreads();

    attn[m * SLEN + s] = a;
    sm[s] = a;
    // coverage loss: sum(min(cov, attn)) / B  (only for t > 0)
    red[s] = (t > 0) ? fminf(cv, a) : 0.0f; __syncthreads();
    for (int off = 64; off > 0; off >>= 1) {
        if (s < off) red[s] += red[s + off];
        __syncthreads();
    }
    if (s == 0 && t > 0) atomicAdd(closs, red[0] * (1.0f / (float)BATCH));
    cov[m * SLEN + s] = cv + a;
    __syncthreads();

    // ctx[f] = sum_s attn[s] * enc[b][s][f]  -> combined[m][COMB/2 + f]
    const float* encb = enc + (size_t)b * SLEN * ENC_H;
    for (int f = s; f < ENC_H; f += SLEN) {
        float acc = 0.0f;
        for (int ss = 0; ss < SLEN; ++ss) acc += sm[ss] * encb[(size_t)ss * ENC_H + f];
        combined[(size_t)m * COMB + HDIM + f] = acc;
    }
}

// ---------------- emotion gate + pointer probability (one block per row) ----------------
__global__ void k_gate_ptr(const float* __restrict__ combined, const float* __restrict__ h_new,
                           const float* __restrict__ emotion,
                           const float* __restrict__ w_gate, const float* __restrict__ b_gate,
                           const float* __restrict__ w_ptr, const float* __restrict__ b_ptr,
                           float* __restrict__ p_ptr) {
    __shared__ float red[D_E];
    const int m = blockIdx.x;
    const int d = threadIdx.x;                 // 256 threads
    const int b = m >> 2;
    const float* cb = combined + (size_t)m * COMB;
    const float* hn = h_new + (size_t)m * HDIM;
    const float* wg = w_gate + (size_t)d * (COMB + HDIM);   // [256, 1536]

    float acc = b_gate[d];
    for (int k = 0; k < COMB; ++k) acc += cb[k] * wg[k];
    for (int k = 0; k < HDIM; ++k) acc += hn[k] * wg[COMB + k];
    float ev = sigf(acc) * emotion[(size_t)b * D_E + d];    // gated emotion vector

    // p_ptr = sigmoid([combined, emo_vec] . w_ptr + b_ptr), w_ptr: [1, COMB + D_E]
    float part = ev * w_ptr[COMB + d];
    for (int k = d * 4; k < d * 4 + 4; ++k) part += cb[k] * w_ptr[k];
    red[d] = part; __syncthreads();
    for (int off = 128; off > 0; off >>= 1) {
        if (d < off) red[d] += red[d + off];
        __syncthreads();
    }
    if (d == 0) p_ptr[m] = sigf(red[0] + b_ptr[0]);
}

// ---------------- vocab softmax + pointer mixture + log-softmax/T ----------------
__global__ void k_dist(float* __restrict__ dist,            // [BB, EXT]; cols 0..VOCAB hold logits on entry
                       const float* __restrict__ p_ptr, const float* __restrict__ attn,
                       const int* __restrict__ ids) {
    __shared__ float red[256];
    const int m = blockIdx.x;
    const int tid = threadIdx.x;               // 256 threads
    const int b = m >> 2;
    float* dm = dist + (size_t)m * EXT;

    float mx = -3.0e38f;
    for (int v = tid; v < VOCAB; v += 256) mx = fmaxf(mx, dm[v]);
    red[tid] = mx; __syncthreads();
    for (int off = 128; off > 0; off >>= 1) {
        if (tid < off) red[tid] = fmaxf(red[tid], red[tid + off]);
        __syncthreads();
    }
    mx = red[0]; __syncthreads();

    float sum = 0.0f;
    for (int v = tid; v < VOCAB; v += 256) sum += expf(dm[v] - mx);
    red[tid] = sum; __syncthreads();
    for (int off = 128; off > 0; off >>= 1) {
        if (tid < off) red[tid] += red[tid + off];
        __syncthreads();
    }
    sum = red[0]; __syncthreads();

    const float p = p_ptr[m];
    const float inv = (1.0f - p) / sum;
    for (int v = tid; v < VOCAB; v += 256) dm[v] = inv * expf(dm[v] - mx);
    for (int v = VOCAB + tid; v < EXT; v += 256) dm[v] = 0.0f;
    __syncthreads();

    for (int s = tid; s < SLEN; s += 256)
        atomicAdd(&dm[ids[b * SLEN + s]], p * attn[m * SLEN + s]);
    __threadfence();
    __syncthreads();

    // log_softmax over already-normalized mixture, then /T
    float s2 = 0.0f;
    for (int v = tid; v < EXT; v += 256) s2 += dm[v];
    red[tid] = s2; __syncthreads();
    for (int off = 128; off > 0; off >>= 1) {
        if (tid < off) red[tid] += red[tid + off];
        __syncthreads();
    }
    const float logZ = logf(red[0]);
    __syncthreads();
    const float scale = 1.0f / (float)TSTEPS;
    for (int v = tid; v < EXT; v += 256)
        dm[v] = (logf(fmaxf(dm[v], 1e-37f)) - logZ) * scale;
}

// ---------------- per-batch top-4 beam step + h reshuffle ----------------
__global__ void k_topk(const float* __restrict__ dist,      // now holds logp
                       float* __restrict__ score, float* __restrict__ scores_last,
                       int* __restrict__ x_tok, int* __restrict__ toks, int* __restrict__ ptrs,
                       const float* __restrict__ h_new, float* __restrict__ h_cur, int t) {
    __shared__ float cv[256 * 4];
    __shared__ int   ci[256 * 4];
    __shared__ int   selBeam[BEAM];
    const int b = blockIdx.x;
    const int tid = threadIdx.x;               // 256 threads

    float tv0 = -3e38f, tv1 = -3e38f, tv2 = -3e38f, tv3 = -3e38f;
    int   ti0 = 0, ti1 = 0, ti2 = 0, ti3 = 0;
    const int total = BEAM * EXT;
    for (int j = tid; j < total; j += 256) {
        const int beam = j / EXT;
        const int v = j - beam * EXT;
        const float val = score[b * BEAM + beam] + dist[(size_t)(b * BEAM + beam) * EXT + v];
        if (val > tv3) {
            if (val > tv0)      { tv3=tv2; ti3=ti2; tv2=tv1; ti2=ti1; tv1=tv0; ti1=ti0; tv0=val; ti0=j; }
            else if (val > tv1) { tv3=tv2; ti3=ti2; tv2=tv1; ti2=ti1; tv1=val; ti1=j; }
            else if (val > tv2) { tv3=tv2; ti3=ti2; tv2=val; ti2=j; }
            else                { tv3=val; ti3=j; }
        }
    }
    cv[tid*4+0]=tv0; ci[tid*4+0]=ti0;  cv[tid*4+1]=tv1; ci[tid*4+1]=ti1;
    cv[tid*4+2]=tv2; ci[tid*4+2]=ti2;  cv[tid*4+3]=tv3; ci[tid*4+3]=ti3;
    __syncthreads();

    if (tid == 0) {
        for (int jj = 0; jj < BEAM; ++jj) {
            int best = 0; float bvv = cv[0];
            for (int q = 1; q < 256 * 4; ++q)
                if (cv[q] > bvv) { bvv = cv[q]; best = q; }
            const int idx = ci[best];
            cv[best] = -3.3e38f;
            const int beam = idx / EXT;
            const int tok  = idx - beam * EXT;
            toks[t * BB + b * BEAM + jj] = tok;
            ptrs[t * BB + b * BEAM + jj] = beam;
            scores_last[b * BEAM + jj] = bvv;
            score[b * BEAM + jj] = (tok == EOS) ? NEGV : bvv;
            x_tok[b * BEAM + jj] = tok;
            selBeam[jj] = beam;
        }
    }
    __syncthreads();
    for (int jj = 0; jj < BEAM; ++jj) {
        const int src = b * BEAM + selBeam[jj];
        const int dst = b * BEAM + jj;
        for (int col = tid; col < HDIM; col += 256)
            h_cur[(size_t)dst * HDIM + col] = h_new[(size_t)src * HDIM + col];
    }
}

// ---------------- backtrack + emit outputs ----------------
__global__ void k_finalize(const int* __restrict__ toks, const int* __restrict__ ptrs,
                           const float* __restrict__ scores_last, const float* __restrict__ closs,
                           float* __restrict__ out) {
    const int tid = threadIdx.x;               // 64 threads, only 32 used for backtrack
    if (tid < BB) {
        const int b = tid / BEAM, j = tid - b * BEAM;
        int p = j;
        for (int tt = TSTEPS - 1; tt >= 0; --tt) {
            out[1 + (size_t)b * BEAM * TSTEPS + (size_t)j * TSTEPS + tt] =
                (float)toks[tt * BB + b * BEAM + p];
            p = ptrs[tt * BB + b * BEAM + p];
        }
        out[1 + BATCH * BEAM * TSTEPS + tid] = scores_last[tid];        // final_score
        out[1 + BATCH * BEAM * TSTEPS + BB + tid] = (float)TSTEPS;      // length
    }
    if (tid == 0) out[0] = closs[0] / (float)TSTEPS;
}

// ---------------- host driver ----------------
extern "C" void kernel_launch(void* const* d_in, const int* in_sizes, int n_in,
                              void* d_out, int out_size, void* d_ws, size_t ws_size,
                              hipStream_t stream) {
    const float* emotion   = (const float*)d_in[0];    // [1,8,256]
    const float* enc       = (const float*)d_in[1];    // [8,128,512]
    const int*   ids       = (const int*)  d_in[2];    // [8,128]
    const float* init_h    = (const float*)d_in[3];    // [1,8,512]
    const float* embedding = (const float*)d_in[5];    // [32000,1024]
    const float* w_ih      = (const float*)d_in[6];    // [1536,1024]
    const float* w_hh      = (const float*)d_in[7];    // [1536,512]
    const float* b_ih      = (const float*)d_in[8];
    const float* b_hh      = (const float*)d_in[9];
    const float* enc_bl_w  = (const float*)d_in[10];   // [512,512]
    const float* enc_bl_b  = (const float*)d_in[11];
    const float* cover_w   = (const float*)d_in[12];
    const float* w_out     = (const float*)d_in[13];   // [32000,1024]
    const float* b_out     = (const float*)d_in[14];
    const float* w_ptr     = (const float*)d_in[15];   // [1,1280]
    const float* b_ptr     = (const float*)d_in[16];
    const float* w_gate    = (const float*)d_in[17];   // [256,1536]
    const float* b_gate    = (const float*)d_in[18];

    // workspace carve-out (256B aligned slices)
    char* wsb = (char*)d_ws;
    size_t off = 0;
    auto alloc = [&](size_t bytes) -> void* {
        void* p = wsb + off;
        off = (off + bytes + 255) & ~(size_t)255;
        return p;
    };
    int*   x_tok       = (int*)  alloc(BB * sizeof(int));
    float* h_cur       = (float*)alloc((size_t)BB * HDIM * 4);
    float* h_new       = (float*)alloc((size_t)BB * HDIM * 4);
    float* emb         = (float*)alloc((size_t)BB * EDIM * 4);
    float* gi          = (float*)alloc((size_t)BB * 3 * HDIM * 4);
    float* gh          = (float*)alloc((size_t)BB * 3 * HDIM * 4);
    float* hW          = (float*)alloc((size_t)BB * ENC_H * 4);
    float* attn        = (float*)alloc((size_t)BB * SLEN * 4);
    float* cov         = (float*)alloc((size_t)BB * SLEN * 4);
    float* combined    = (float*)alloc((size_t)BB * COMB * 4);
    float* p_ptr       = (float*)alloc(BB * 4);
    float* score       = (float*)alloc(BB * 4);
    float* scores_last = (float*)alloc(BB * 4);
    float* closs       = (float*)alloc(256);
    float* dist        = (float*)alloc((size_t)BB * EXT * 4);
    int*   toks        = (int*)  alloc((size_t)TSTEPS * BB * sizeof(int));
    int*   ptrs        = (int*)  alloc((size_t)TSTEPS * BB * sizeof(int));
    (void)ws_size; (void)in_sizes; (void)n_in; (void)out_size;

    k_init<<<64, 256, 0, stream>>>(h_cur, cov, x_tok, score, closs, init_h);

    for (int t = 0; t < TSTEPS; ++t) {
        k_embed<<<(BB * EDIM + 255) / 256, 256, 0, stream>>>(x_tok, embedding, emb);

        // gi = emb @ w_ih^T + b_ih   : M=32, N=1536, K=1024
        k_gemm_wmma<true><<<((1536 / 16) * 2 + 7) / 8, 256, 0, stream>>>(
            emb, w_ih, b_ih, gi, 3 * HDIM, EDIM, EDIM, EDIM, 3 * HDIM);
        // gh = h @ w_hh^T + b_hh     : M=32, N=1536, K=512
        k_gemm_wmma<true><<<((1536 / 16) * 2 + 7) / 8, 256, 0, stream>>>(
            h_cur, w_hh, b_hh, gh, 3 * HDIM, HDIM, HDIM, HDIM, 3 * HDIM);

        k_gru_combine<<<(BB * HDIM + 255) / 256, 256, 0, stream>>>(gi, gh, h_cur, h_new, combined);

        // hW = h_new @ enc_bl_w      : M=32, N=512, K=512 (B not transposed)
        k_gemm_wmma<false><<<((ENC_H / 16) * 2 + 7) / 8, 256, 0, stream>>>(
            h_new, enc_bl_w, nullptr, hW, ENC_H, HDIM, HDIM, ENC_H, ENC_H);

        k_attn<<<BB, SLEN, 0, stream>>>(hW, enc, enc_bl_b, cover_w, cov, attn, combined, closs, t);

        k_gate_ptr<<<BB, D_E, 0, stream>>>(combined, h_new, emotion, w_gate, b_gate,
                                           w_ptr, b_ptr, p_ptr);

        // logits = combined @ w_out^T + b_out : M=32, N=32000, K=1024 -> into dist (ldd=EXT)
        k_gemm_wmma<true><<<((VOCAB / 16) * 2 + 7) / 8, 256, 0, stream>>>(
            combined, w_out, b_out, dist, VOCAB, COMB, COMB, COMB, EXT);

        k_dist<<<BB, 256, 0, stream>>>(dist, p_ptr, attn, ids);

        k_topk<<<BATCH, 256, 0, stream>>>(dist, score, scores_last, x_tok, toks, ptrs,
                                          h_new, h_cur, t);
    }

    k_finalize<<<1, 64, 0, stream>>>(toks, ptrs, scores_last, closs, (float*)d_out);
}